// SparseMoELayer_5583457485161
// MI455X (gfx1250) — compile-verified
//
#include <hip/hip_runtime.h>
#include <math.h>

// ---------------- problem constants ----------------
constexpr int kB    = 4;
constexpr int kC    = 64;          // channels
constexpr int kH    = 128;
constexpr int kW    = 128;
constexpr int kE    = 32;          // experts
constexpr int kHW   = kH * kW;               // 16384
constexpr int kCHW  = kC * kHW;              // 1048576
constexpr int kNPix = kB * kHW;              // 65536
constexpr int kOutN = kB * kCHW;             // 4194304
constexpr int kKTot = kC * 9;                // 576
constexpr int kWPerE = kC * kKTot;           // 36864 floats per expert
constexpr int kSortCap = 2 * kNPix + kE * 16;   // 131584
constexpr int kNGroup  = kSortCap / 16;          // 8224
constexpr int kWtN  = kE * 9 * kC * kC;      // 1179648 transposed weights

// ---------------- workspace layout (int units) ----------------
constexpr int WS_TOPIDX = 0;                       // 2*kNPix ints
constexpr int WS_TOPW   = 2 * kNPix;               // 2*kNPix floats
constexpr int WS_CNT    = 4 * kNPix;               // 32 ints
constexpr int WS_OFF    = WS_CNT + 32;             // 33 ints
constexpr int WS_CUR    = WS_OFF + 40;             // 32 ints
constexpr int WS_SPIX   = WS_CUR + 32;             // kSortCap ints
constexpr int WS_SW     = WS_SPIX + kSortCap;      // kSortCap floats
constexpr int WS_WT     = WS_SW + kSortCap;        // kWtN floats: [e][tap][cout][cin]

typedef float v2f __attribute__((ext_vector_type(2)));
typedef float v8f __attribute__((ext_vector_type(8)));

// ---------------- kernel 0: weight transpose [e][cout][cin][tap] -> [e][tap][cout][cin] ----------------
__global__ void moe_wtr(const float* __restrict__ expert_w, int* __restrict__ ws) {
    int i = blockIdx.x * blockDim.x + threadIdx.x;
    if (i >= kWtN) return;
    int cin  = i & 63;
    int cout = (i >> 6) & 63;
    int et   = i >> 12;            // e*9 + tap
    int tap  = et % 9;
    int e    = et / 9;
    ((float*)(ws + WS_WT))[i] =
        expert_w[(((e * kC + cout) * kC + cin) * 9) + tap];
}

// ---------------- kernel 1: init accumulator / counters / pad markers ----------------
__global__ void moe_init(float* __restrict__ out, int* __restrict__ ws) {
    int i = blockIdx.x * blockDim.x + threadIdx.x;
    if (i < kOutN)    out[i] = 0.0f;
    if (i < kSortCap) ws[WS_SPIX + i] = -1;
    if (i < kE)       ws[WS_CNT + i] = 0;
}

// ---------------- kernel 2: gate (1x1 conv) + top-2 + softmax + counts ----------------
__global__ void moe_gate(const float* __restrict__ x,
                         const float* __restrict__ gate_w,
                         const float* __restrict__ gate_b,
                         int* __restrict__ ws) {
    int p = blockIdx.x * blockDim.x + threadIdx.x;
    if (p >= kNPix) return;
    int b  = p >> 14;
    int hw = p & (kHW - 1);
    const float* xb = x + b * kCHW + hw;

    float xv[kC];
#pragma unroll
    for (int c = 0; c < kC; ++c) xv[c] = xb[c * kHW];

    float v0 = -INFINITY, v1 = -INFINITY;
    int   i0 = 0, i1 = 0;
    for (int e = 0; e < kE; ++e) {
        float acc = gate_b[e];
        const float* gw = gate_w + e * kC;
#pragma unroll
        for (int c = 0; c < kC; ++c) acc = fmaf(xv[c], gw[c], acc);
        if (acc > v0)      { v1 = v0; i1 = i0; v0 = acc; i0 = e; }
        else if (acc > v1) { v1 = acc; i1 = e; }
    }
    float e1 = expf(v1 - v0);
    float inv = 1.0f / (1.0f + e1);

    ws[WS_TOPIDX + 2 * p]     = i0;
    ws[WS_TOPIDX + 2 * p + 1] = i1;
    float* topw = (float*)(ws + WS_TOPW);
    topw[2 * p]     = inv;
    topw[2 * p + 1] = e1 * inv;
    atomicAdd(&ws[WS_CNT + i0], 1);
    atomicAdd(&ws[WS_CNT + i1], 1);
}

// ---------------- kernel 3: exclusive scan, each expert segment padded to 16 ----------------
__global__ void moe_prefix(int* __restrict__ ws) {
    if (threadIdx.x == 0) {
        int run = 0;
        for (int e = 0; e < kE; ++e) {
            ws[WS_OFF + e] = run;
            ws[WS_CUR + e] = run;
            run += (ws[WS_CNT + e] + 15) & ~15;
        }
        ws[WS_OFF + kE] = run;
    }
}

// ---------------- kernel 4: counting-sort scatter of (pixel, weight) by expert ----------------
__global__ void moe_scatter(int* __restrict__ ws) {
    int p = blockIdx.x * blockDim.x + threadIdx.x;
    if (p >= kNPix) return;
    float* topw = (float*)(ws + WS_TOPW);
    float* sw   = (float*)(ws + WS_SW);
#pragma unroll
    for (int k = 0; k < 2; ++k) {
        int e   = ws[WS_TOPIDX + 2 * p + k];
        int pos = atomicAdd(&ws[WS_CUR + e], 1);
        ws[WS_SPIX + pos] = p;
        sw[pos] = topw[2 * p + k];
    }
}

// ---------------- kernel 5: sparse expert 3x3 conv, implicit GEMM via f32 WMMA ----------------
// One wave per group of 16 same-expert assignments.
// M=16 pixels, N=64 couts (4 tiles), K ordered tap-outer (9, unrolled) x cin-inner (64).
__global__ void __launch_bounds__(32)
moe_conv(const float* __restrict__ x,
         const float* __restrict__ expert_b,
         const int* __restrict__ ws,
         float* __restrict__ out) {
    int g16 = blockIdx.x * 16;
    if (g16 >= ws[WS_OFF + kE]) return;

    int e = 0;
    for (int t = 0; t < kE; ++t) {
        if (g16 >= ws[WS_OFF + t] && g16 < ws[WS_OFF + t + 1]) { e = t; break; }
    }

    int lane = threadIdx.x;
    __shared__ int   sPix[16];
    __shared__ float sWgt[16];
    if (lane < 16) {
        sPix[lane] = ws[WS_SPIX + g16 + lane];
        sWgt[lane] = ((const float*)(ws + WS_SW))[g16 + lane];
    }
    __syncthreads();

    // A-frag addressing: lane L -> row M = L&15, K pair = k + (L>>4)*2 + {0,1}
    int pA  = sPix[lane & 15];
    bool av = (pA >= 0);
    int ah  = (pA >> 7) & 127;
    int aw  = pA & 127;
    const float* xb = x + ((pA >> 14) & 3) * kCHW;

    // transposed weights for this expert: [tap][cout][cin]
    const float* wt = ((const float*)(ws + WS_WT)) + e * (9 * kC * kC);
    __builtin_prefetch(wt + lane * (9 * kC * kC / 32), 0, 1);   // global_prefetch_b8

    int n     = lane & 15;          // B col within tile
    int klane = (lane >> 4) << 1;   // 0 or 2

    v8f acc0 = {}, acc1 = {}, acc2 = {}, acc3 = {};

#pragma unroll
    for (int tap = 0; tap < 9; ++tap) {
        const int dy = tap / 3 - 1;     // constant after unroll
        const int dx = tap % 3 - 1;
        int hh = ah + dy, ww = aw + dx;
        bool ok = av & (hh >= 0) & (hh < kH) & (ww >= 0) & (ww < kW);
        const float* xa  = xb + (ok ? (hh * kW + ww) : 0);   // safe clamped base
        const float* wtt = wt + tap * (kC * kC) + n * kC;

#pragma unroll 4
        for (int c0 = 0; c0 < kC; c0 += 4) {
            int kk = c0 + klane;
            float r0 = xa[kk * kHW];
            float r1 = xa[(kk + 1) * kHW];
            v2f a;
            a.x = ok ? r0 : 0.0f;
            a.y = ok ? r1 : 0.0f;
            v2f b0 = *(const v2f*)(wtt + kk);
            v2f b1 = *(const v2f*)(wtt + 16 * kC + kk);
            v2f b2 = *(const v2f*)(wtt + 32 * kC + kk);
            v2f b3 = *(const v2f*)(wtt + 48 * kC + kk);

            acc0 = __builtin_amdgcn_wmma_f32_16x16x4_f32(false, a, false, b0, (short)0, acc0, false, false);
            acc1 = __builtin_amdgcn_wmma_f32_16x16x4_f32(false, a, false, b1, (short)0, acc1, false, false);
            acc2 = __builtin_amdgcn_wmma_f32_16x16x4_f32(false, a, false, b2, (short)0, acc2, false, false);
            acc3 = __builtin_amdgcn_wmma_f32_16x16x4_f32(false, a, false, b3, (short)0, acc3, false, false);
        }
    }

    // epilogue: lane j, vgpr r -> pixel m = r + 8*(j>=16), cout = (j&15) + 16*t
    int mofs = (lane >> 4) << 3;
    const float* bias = expert_b + e * kC;
#pragma unroll
    for (int t = 0; t < 4; ++t) {
        int cout = n + 16 * t;
        float bv = bias[cout];
        v8f acc = (t == 0) ? acc0 : (t == 1) ? acc1 : (t == 2) ? acc2 : acc3;
#pragma unroll
        for (int r = 0; r < 8; ++r) {
            int m = r + mofs;
            int p = sPix[m];
            if (p >= 0) {
                float val = sWgt[m] * (acc[r] + bv);
                unsafeAtomicAdd(&out[((p >> 14) & 3) * kCHW + cout * kHW + (p & (kHW - 1))], val);
            }
        }
    }
}

// ---------------- kernel 6: in-place ReLU ----------------
__global__ void moe_relu(float* __restrict__ out) {
    int i = blockIdx.x * blockDim.x + threadIdx.x;
    if (i < kOutN) out[i] = fmaxf(out[i], 0.0f);
}

// ---------------- launch ----------------
extern "C" void kernel_launch(void* const* d_in, const int* in_sizes, int n_in,
                              void* d_out, int out_size, void* d_ws, size_t ws_size,
                              hipStream_t stream) {
    const float* x        = (const float*)d_in[0];  // [4,64,128,128]
    const float* expert_w = (const float*)d_in[1];  // [32,64,64,3,3]
    const float* expert_b = (const float*)d_in[2];  // [32,64]
    const float* gate_w   = (const float*)d_in[3];  // [32,64]
    const float* gate_b   = (const float*)d_in[4];  // [32]
    float* out = (float*)d_out;
    int*   ws  = (int*)d_ws;

    moe_wtr    <<<(kWtN + 255) / 256, 256, 0, stream>>>(expert_w, ws);
    moe_init   <<<(kOutN + 255) / 256, 256, 0, stream>>>(out, ws);
    moe_gate   <<<(kNPix + 255) / 256, 256, 0, stream>>>(x, gate_w, gate_b, ws);
    moe_prefix <<<1, 32, 0, stream>>>(ws);
    moe_scatter<<<(kNPix + 255) / 256, 256, 0, stream>>>(ws);
    moe_conv   <<<kNGroup, 32, 0, stream>>>(x, expert_b, ws, out);
    moe_relu   <<<(kOutN + 255) / 256, 256, 0, stream>>>(out);
}